// GATLayer_55490977464421
// MI455X (gfx1250) — compile-verified
//
#include <hip/hip_runtime.h>
#include <stdint.h>

#define HEADS 4
#define OUT_DIM 32
#define IN_DIM 128   // == HEADS*OUT_DIM

typedef __attribute__((ext_vector_type(2))) float v2f;
typedef __attribute__((ext_vector_type(8))) float v8f;

// ---------- helpers ----------
__device__ __forceinline__ unsigned f2k(float f) {
    unsigned u = __float_as_uint(f);
    return (u & 0x80000000u) ? ~u : (u | 0x80000000u);   // monotonic float->uint key
}
__device__ __forceinline__ float k2f(unsigned k) {
    unsigned u = (k & 0x80000000u) ? (k ^ 0x80000000u) : ~k;
    return __uint_as_float(u);
}
// logical edge list = [ (s0,d0) | (d0,s0) | self-loops ]
__device__ __forceinline__ void edge_sd(const int* __restrict__ ei, int E, int e,
                                        int& s, int& d) {
    if (e < E)            { s = ei[e];     d = ei[e + E]; }
    else if (e < 2 * E)   { s = ei[e];     d = ei[e - E]; }   // ei[e]=d0[e-E]
    else                  { s = e - 2 * E; d = s; }
}

// ---------- kernels ----------
__global__ void init_kernel(float* __restrict__ out, unsigned* __restrict__ keys,
                            float* __restrict__ denom, int n_out, int n_nh) {
    int i = blockIdx.x * blockDim.x + threadIdx.x;
    if (i < n_out) out[i] = 0.0f;
    if (i < n_nh)  { keys[i] = 0u; denom[i] = 0.0f; }
}

// proj[N,128] = x[N,128] @ W[128,128], fp32 WMMA 16x16x4.
// One wave per 16x16 tile; block = 8 waves = the 8 N-tiles of one M-tile.
__global__ __launch_bounds__(256) void gemm_wmma(const float* __restrict__ x,
                                                 const float* __restrict__ W,
                                                 float* __restrict__ proj, int N) {
    const int lane = threadIdx.x & 31;
    const int wave = threadIdx.x >> 5;     // N-tile 0..7
    const int m0 = blockIdx.x * 16;
    if (m0 >= N) return;
    const int n0 = wave * 16;
    const int r  = lane & 15;              // A: M row / B,D: N col
    const int kh = lane >> 4;              // K half-select (0: K0/K1, 1: K2/K3)

    const float* __restrict__ xrow = x + (size_t)(m0 + r) * IN_DIM;
    v8f acc = {};
#pragma unroll 4
    for (int kb = 0; kb < IN_DIM / 4; ++kb) {
        const int k = kb * 4 + kh * 2;
        v2f a, b;
        a.x = xrow[k];
        a.y = xrow[k + 1];
        b.x = W[(size_t)k       * IN_DIM + n0 + r];
        b.y = W[(size_t)(k + 1) * IN_DIM + n0 + r];
        acc = __builtin_amdgcn_wmma_f32_16x16x4_f32(
            /*neg_a=*/false, a, /*neg_b=*/false, b,
            /*c_mod=*/(short)0, acc, /*reuse_a=*/false, /*reuse_b=*/false);
    }
    // D layout: VGPR i -> M = i + 8*(lane/16), N = lane%16
    const int col = n0 + r;
#pragma unroll
    for (int i = 0; i < 8; ++i) {
        const int row = m0 + kh * 8 + i;
        proj[(size_t)row * IN_DIM + col] = acc[i];
    }
}

// Per-node score components: s_src[n,h]=dot(proj[n,h,:],attn_src[h,:]) (same for dst).
// One wave per node; lane l covers head l>>3, elems ((l&7)*4 .. +3).
__global__ __launch_bounds__(256) void node_scores(const float* __restrict__ proj,
                                                   const float* __restrict__ asrc,
                                                   const float* __restrict__ adst,
                                                   float* __restrict__ ssrc,
                                                   float* __restrict__ sdst, int N) {
    const int node = (blockIdx.x * blockDim.x + threadIdx.x) >> 5;
    const int lane = threadIdx.x & 31;
    if (node >= N) return;
    const int h = lane >> 3;
    const int j = (lane & 7) * 4;
    const float4 p  = *(const float4*)(proj + (size_t)node * IN_DIM + h * OUT_DIM + j);
    const float4 as = *(const float4*)(asrc + h * OUT_DIM + j);
    const float4 ad = *(const float4*)(adst + h * OUT_DIM + j);
    float s = p.x * as.x + p.y * as.y + p.z * as.z + p.w * as.w;
    float d = p.x * ad.x + p.y * ad.y + p.z * ad.z + p.w * ad.w;
#pragma unroll
    for (int m = 1; m < 8; m <<= 1) {
        s += __shfl_xor(s, m, 32);
        d += __shfl_xor(d, m, 32);
    }
    if ((lane & 7) == 0) {
        ssrc[node * HEADS + h] = s;
        sdst[node * HEADS + h] = d;
    }
}

// Pass 1: segment max via monotonic-key atomic umax. One thread per (edge,head).
__global__ void edge_max(const int* __restrict__ ei, const float* __restrict__ ssrc,
                         const float* __restrict__ sdst, unsigned* __restrict__ keys,
                         int E, int EaH) {
    int idx = blockIdx.x * blockDim.x + threadIdx.x;
    if (idx >= EaH) return;
    const int e = idx >> 2, h = idx & 3;
    int s, d; edge_sd(ei, E, e, s, d);
    const float t  = ssrc[s * HEADS + h] + sdst[d * HEADS + h];
    const float sc = t > 0.0f ? t : 0.2f * t;              // leaky_relu(0.2)
    atomicMax(&keys[d * HEADS + h], f2k(sc));
}

// Pass 2: denom[n,h] = sum exp(score - max)
__global__ void edge_denom(const int* __restrict__ ei, const float* __restrict__ ssrc,
                           const float* __restrict__ sdst, const unsigned* __restrict__ keys,
                           float* __restrict__ denom, int E, int EaH) {
    int idx = blockIdx.x * blockDim.x + threadIdx.x;
    if (idx >= EaH) return;
    const int e = idx >> 2, h = idx & 3;
    int s, d; edge_sd(ei, E, e, s, d);
    const float t  = ssrc[s * HEADS + h] + sdst[d * HEADS + h];
    const float sc = t > 0.0f ? t : 0.2f * t;
    const float m  = k2f(keys[d * HEADS + h]);
    atomicAdd(&denom[d * HEADS + h], __expf(sc - m));
}

// Pass 3: out[dst] += attn * proj[src]. One wave per edge; lane covers 4 floats.
__global__ __launch_bounds__(256) void edge_msg(const int* __restrict__ ei,
                                                const float* __restrict__ proj,
                                                const float* __restrict__ ssrc,
                                                const float* __restrict__ sdst,
                                                const unsigned* __restrict__ keys,
                                                const float* __restrict__ denom,
                                                float* __restrict__ out, int E, int Ea) {
    const int e    = (blockIdx.x * blockDim.x + threadIdx.x) >> 5;
    const int lane = threadIdx.x & 31;
    if (e >= Ea) return;
    int s, d; edge_sd(ei, E, e, s, d);
    const int h = lane >> 3;
    const float t  = ssrc[s * HEADS + h] + sdst[d * HEADS + h];
    const float sc = t > 0.0f ? t : 0.2f * t;
    const float m  = k2f(keys[d * HEADS + h]);
    const float dn = fmaxf(denom[d * HEADS + h], 1e-12f);
    const float w  = __expf(sc - m) / dn;
    const int base = h * OUT_DIM + (lane & 7) * 4;
    const float4 p = *(const float4*)(proj + (size_t)s * IN_DIM + base);
    float* o = out + (size_t)d * IN_DIM + base;
    atomicAdd(o + 0, w * p.x);
    atomicAdd(o + 1, w * p.y);
    atomicAdd(o + 2, w * p.z);
    atomicAdd(o + 3, w * p.w);
}

// out = elu(out + bias)
__global__ void finalize(float* __restrict__ out, const float* __restrict__ bias, int total) {
    int i = blockIdx.x * blockDim.x + threadIdx.x;
    if (i >= total) return;
    const float v = out[i] + bias[i & (IN_DIM - 1)];
    out[i] = v > 0.0f ? v : expm1f(v);
}

// ---------- launch ----------
extern "C" void kernel_launch(void* const* d_in, const int* in_sizes, int n_in,
                              void* d_out, int out_size, void* d_ws, size_t ws_size,
                              hipStream_t stream) {
    const float* x    = (const float*)d_in[0];
    const int*   ei   = (const int*)d_in[1];
    const float* W    = (const float*)d_in[2];
    const float* asrc = (const float*)d_in[3];
    const float* adst = (const float*)d_in[4];
    const float* bias = (const float*)d_in[5];

    const int N  = in_sizes[0] / IN_DIM;
    const int E  = in_sizes[1] / 2;
    const int Ea = 2 * E + N;

    float*    proj  = (float*)d_ws;
    float*    ssrc  = proj + (size_t)N * IN_DIM;
    float*    sdst  = ssrc + (size_t)N * HEADS;
    unsigned* keys  = (unsigned*)(sdst + (size_t)N * HEADS);
    float*    denom = (float*)(keys + (size_t)N * HEADS);

    float* out = (float*)d_out;
    const int nOut = N * IN_DIM;
    const int nNH  = N * HEADS;
    const int EaH  = Ea * HEADS;

    init_kernel<<<(nOut + 255) / 256, 256, 0, stream>>>(out, keys, denom, nOut, nNH);
    gemm_wmma<<<(N + 15) / 16, 256, 0, stream>>>(x, W, proj, N);
    node_scores<<<(N + 7) / 8, 256, 0, stream>>>(proj, asrc, adst, ssrc, sdst, N);
    edge_max<<<(EaH + 255) / 256, 256, 0, stream>>>(ei, ssrc, sdst, keys, E, EaH);
    edge_denom<<<(EaH + 255) / 256, 256, 0, stream>>>(ei, ssrc, sdst, keys, denom, E, EaH);
    edge_msg<<<(Ea + 7) / 8, 256, 0, stream>>>(ei, proj, ssrc, sdst, keys, denom, out, E, Ea);
    finalize<<<(nOut + 255) / 256, 256, 0, stream>>>(out, bias, nOut);
}